// CurveRenderwithLUT_40484361732650
// MI455X (gfx1250) — compile-verified
//
#include <hip/hip_runtime.h>

#define DIMLUT 33
#define NLUT   (DIMLUT * DIMLUT * DIMLUT)   // 35937
#define HW     262144                        // 512*512
#define NPIX   4194304                       // 16*512*512
#define OUT_IMG 12582912                     // 16*3*512*512

typedef __attribute__((ext_vector_type(2))) float v2f;
typedef __attribute__((ext_vector_type(8))) float v8f;

__device__ __forceinline__ float fast_tanh(float v) {
#if __has_builtin(__builtin_amdgcn_tanhf)
    return __builtin_amdgcn_tanhf(v);       // v_tanh_f32 on gfx1250
#else
    return tanhf(v);
#endif
}

// ---------------------------------------------------------------------------
// Kernel 1: param[16,3] = b_feature@Wb.T + bb + f_feature@Wf.T + bf
// via V_WMMA_F32_16X16X4_F32, one wave.  M=16 rows = batch, N cols 0..2 valid
// (cols >= 3 masked to zero so EXEC stays all-ones and loads stay in-bounds).
// ---------------------------------------------------------------------------
__global__ __launch_bounds__(32) void param_wmma_kernel(
    const float* __restrict__ f_feat, const float* __restrict__ b_feat,
    const float* __restrict__ Wf, const float* __restrict__ bf,
    const float* __restrict__ Wb, const float* __restrict__ bb,
    float* __restrict__ param_out)
{
    const int lane = threadIdx.x;            // 0..31
    const int m    = lane & 15;              // A row (batch), B col
    const int koff = (lane >> 4) << 1;       // 0 or 2
    const float bmask = (m < 3) ? 1.0f : 0.0f;
    const int   nn    = (m < 3) ? m : 0;     // clamp to stay in-bounds

    const float* __restrict__ fA = f_feat + m  * 512;
    const float* __restrict__ bA = b_feat + m  * 512;
    const float* __restrict__ fB = Wf     + nn * 512;
    const float* __restrict__ bB = Wb     + nn * 512;

    v8f acc = {};
#pragma unroll 4
    for (int k0 = 0; k0 < 512; k0 += 4) {
        v2f a, b;
        a.x = fA[k0 + koff];            a.y = fA[k0 + koff + 1];
        b.x = fB[k0 + koff] * bmask;    b.y = fB[k0 + koff + 1] * bmask;
        acc = __builtin_amdgcn_wmma_f32_16x16x4_f32(false, a, false, b,
                                                    (short)0, acc, false, false);
        a.x = bA[k0 + koff];            a.y = bA[k0 + koff + 1];
        b.x = bB[k0 + koff] * bmask;    b.y = bB[k0 + koff + 1] * bmask;
        acc = __builtin_amdgcn_wmma_f32_16x16x4_f32(false, a, false, b,
                                                    (short)0, acc, false, false);
    }

    const int col   = lane & 15;
    const int rbase = (lane >> 4) * 8;
    if (col < 3) {
        const float bias = bf[col] + bb[col];
#pragma unroll
        for (int v = 0; v < 8; ++v)
            param_out[(rbase + v) * 3 + col] = acc[v] + bias;
    }
}

// ---------------------------------------------------------------------------
// Kernel 2: repack the 3 LUTs (each [3,33,33,33]) into AoS entries of 12
// floats (48 B, 16B aligned): [l0c0 l0c1 l0c2 l1c0 l1c1 l1c2 l2c0 l2c1 l2c2 0 0 0]
// so the main kernel gathers 3x b128 per corner instead of 9 scalar dwords.
// ---------------------------------------------------------------------------
__global__ __launch_bounds__(256) void repack_lut_kernel(
    const float* __restrict__ lut0, const float* __restrict__ lut1,
    const float* __restrict__ lut2, float* __restrict__ wsl)
{
    int i = blockIdx.x * blockDim.x + threadIdx.x;
    if (i >= NLUT) return;
    float4* o = (float4*)(wsl + (size_t)i * 12);
    float4 e0, e1, e2;
    e0.x = lut0[i]; e0.y = lut0[NLUT + i]; e0.z = lut0[2 * NLUT + i];
    e0.w = lut1[i]; e1.x = lut1[NLUT + i]; e1.y = lut1[2 * NLUT + i];
    e1.z = lut2[i]; e1.w = lut2[NLUT + i]; e2.x = lut2[2 * NLUT + i];
    e2.y = 0.0f; e2.z = 0.0f; e2.w = 0.0f;
    o[0] = e0; o[1] = e1; o[2] = e2;
}

// ---------------------------------------------------------------------------
// Kernel 3: per-pixel tanh + 3x trilinear LUT + blend + multiply-by-x.
// Streaming x/out accesses use non-temporal hints (TH=NT) so the one-touch
// 100MB stream does not evict LUT lines from WGP$/L2.
// ---------------------------------------------------------------------------
__device__ __forceinline__ void trilinear_setup(float xr, float xg, float xb,
                                                int& lin, float* w)
{
    const float invbin = 32.0f / 1.000001f;   // (dim-1)/1.000001
    float pr = fast_tanh(xr) * invbin;
    float pg = fast_tanh(xg) * invbin;
    float pb = fast_tanh(xb) * invbin;
    int rid = min(max((int)floorf(pr), 0), 31);
    int gid = min(max((int)floorf(pg), 0), 31);
    int bid = min(max((int)floorf(pb), 0), 31);
    float rd = pr - (float)rid, gd = pg - (float)gid, bd = pb - (float)bid;
    float r0 = 1.0f - rd, g0 = 1.0f - gd, b0 = 1.0f - bd;
    lin = rid + gid * DIMLUT + bid * (DIMLUT * DIMLUT);
    w[0] = r0 * g0 * b0;  w[1] = rd * g0 * b0;
    w[2] = r0 * gd * b0;  w[3] = rd * gd * b0;
    w[4] = r0 * g0 * bd;  w[5] = rd * g0 * bd;
    w[6] = r0 * gd * bd;  w[7] = rd * gd * bd;
}

__constant__ const int c_offs[8] = {0, 1, 33, 34, 1089, 1090, 1122, 1123};

__global__ __launch_bounds__(256) void lut_apply_packed_kernel(
    const float* __restrict__ x, const float* __restrict__ param,
    const float* __restrict__ wsl, float* __restrict__ out)
{
    int p = blockIdx.x * blockDim.x + threadIdx.x;
    if (p >= NPIX) return;
    int b  = p >> 18;              // p / (512*512)
    int hw = p & (HW - 1);

    const float* xb = x + (size_t)b * 3 * HW + hw;
    float xr  = __builtin_nontemporal_load(xb);
    float xg  = __builtin_nontemporal_load(xb + HW);
    float xbl = __builtin_nontemporal_load(xb + 2 * HW);

    int lin; float w[8];
    trilinear_setup(xr, xg, xbl, lin, w);

    float acc[9] = {0, 0, 0, 0, 0, 0, 0, 0, 0};
#pragma unroll
    for (int c = 0; c < 8; ++c) {
        const float4* e = (const float4*)(wsl + (size_t)(lin + c_offs[c]) * 12);
        float4 e0 = e[0], e1 = e[1], e2 = e[2];
        float wc = w[c];
        acc[0] += wc * e0.x; acc[1] += wc * e0.y; acc[2] += wc * e0.z;
        acc[3] += wc * e0.w; acc[4] += wc * e1.x; acc[5] += wc * e1.y;
        acc[6] += wc * e1.z; acc[7] += wc * e1.w; acc[8] += wc * e2.x;
    }

    float p0 = param[b * 3 + 0], p1 = param[b * 3 + 1], p2 = param[b * 3 + 2];
    float* ob = out + (size_t)b * 3 * HW + hw;
    __builtin_nontemporal_store((p0 * acc[0] + p1 * acc[3] + p2 * acc[6]) * xr,  ob);
    __builtin_nontemporal_store((p0 * acc[1] + p1 * acc[4] + p2 * acc[7]) * xg,  ob + HW);
    __builtin_nontemporal_store((p0 * acc[2] + p1 * acc[5] + p2 * acc[8]) * xbl, ob + 2 * HW);
}

// Fallback: direct gathers from the original LUT layout (no scratch needed).
__global__ __launch_bounds__(256) void lut_apply_direct_kernel(
    const float* __restrict__ x, const float* __restrict__ param,
    const float* __restrict__ lut0, const float* __restrict__ lut1,
    const float* __restrict__ lut2, float* __restrict__ out)
{
    int p = blockIdx.x * blockDim.x + threadIdx.x;
    if (p >= NPIX) return;
    int b  = p >> 18;
    int hw = p & (HW - 1);

    const float* xb = x + (size_t)b * 3 * HW + hw;
    float xr  = __builtin_nontemporal_load(xb);
    float xg  = __builtin_nontemporal_load(xb + HW);
    float xbl = __builtin_nontemporal_load(xb + 2 * HW);

    int lin; float w[8];
    trilinear_setup(xr, xg, xbl, lin, w);

    float acc[9] = {0, 0, 0, 0, 0, 0, 0, 0, 0};
#pragma unroll
    for (int c = 0; c < 8; ++c) {
        int l = lin + c_offs[c];
        float wc = w[c];
        acc[0] += wc * lut0[l];            acc[1] += wc * lut0[NLUT + l];
        acc[2] += wc * lut0[2 * NLUT + l];
        acc[3] += wc * lut1[l];            acc[4] += wc * lut1[NLUT + l];
        acc[5] += wc * lut1[2 * NLUT + l];
        acc[6] += wc * lut2[l];            acc[7] += wc * lut2[NLUT + l];
        acc[8] += wc * lut2[2 * NLUT + l];
    }

    float p0 = param[b * 3 + 0], p1 = param[b * 3 + 1], p2 = param[b * 3 + 2];
    float* ob = out + (size_t)b * 3 * HW + hw;
    __builtin_nontemporal_store((p0 * acc[0] + p1 * acc[3] + p2 * acc[6]) * xr,  ob);
    __builtin_nontemporal_store((p0 * acc[1] + p1 * acc[4] + p2 * acc[7]) * xg,  ob + HW);
    __builtin_nontemporal_store((p0 * acc[2] + p1 * acc[5] + p2 * acc[8]) * xbl, ob + 2 * HW);
}

extern "C" void kernel_launch(void* const* d_in, const int* in_sizes, int n_in,
                              void* d_out, int out_size, void* d_ws, size_t ws_size,
                              hipStream_t stream)
{
    (void)in_sizes; (void)n_in; (void)out_size;
    const float* x         = (const float*)d_in[0];
    const float* f_feature = (const float*)d_in[1];
    const float* b_feature = (const float*)d_in[2];
    const float* Wf        = (const float*)d_in[3];
    const float* bf        = (const float*)d_in[4];
    const float* Wb        = (const float*)d_in[5];
    const float* bb        = (const float*)d_in[6];
    const float* lut0      = (const float*)d_in[7];
    const float* lut1      = (const float*)d_in[8];
    const float* lut2      = (const float*)d_in[9];

    float* out       = (float*)d_out;
    float* param_out = out + OUT_IMG;     // second tuple element: param [16,3]

    // 1) param via WMMA (one wave)
    param_wmma_kernel<<<1, 32, 0, stream>>>(f_feature, b_feature, Wf, bf, Wb, bb,
                                            param_out);

    const size_t need = (size_t)NLUT * 12 * sizeof(float);
    if (ws_size >= need) {
        float* wsl = (float*)d_ws;
        // 2) repack LUTs into 48B AoS entries
        repack_lut_kernel<<<(NLUT + 255) / 256, 256, 0, stream>>>(lut0, lut1, lut2,
                                                                  wsl);
        // 3) main pass: b128 gathers
        lut_apply_packed_kernel<<<NPIX / 256, 256, 0, stream>>>(x, param_out, wsl,
                                                                out);
    } else {
        lut_apply_direct_kernel<<<NPIX / 256, 256, 0, stream>>>(x, param_out, lut0,
                                                                lut1, lut2, out);
    }
}